// CustomMSELoss_50113678410134
// MI455X (gfx1250) — compile-verified
//
#include <hip/hip_runtime.h>

#define TPB 256
#define NBLK 2048

// ---- workspace layout (32-bit words) ----
#define OFF_H1    0                  // 2048  u32 : pass-1 histogram (key[31:21])
#define OFF_H2    2048               // 4*2048 u32: pass-2 histograms (key[20:10]) per rank
#define OFF_H3    10240              // 4*1024 u32: pass-3 histograms (key[9:0]) per rank
#define OFF_SEL1  14336              // 4 u32
#define OFF_REM1  14340              // 4 u32
#define OFF_SEL2  14344              // 4 u32
#define OFF_REM2  14348              // 4 u32
#define OFF_VALS  14352              // 4 f32 (order statistics)
#define OFF_THR   14356              // 2 f32 (tL, tR)
#define OFF_SUMP  14360              // NBLK f32
#define OFF_SUMP2 (14360 + NBLK)     // NBLK f32
#define OFF_MSE   (14360 + 2*NBLK)   // NBLK f32
#define ZERO_WORDS 14336             // only histogram region needs zeroing

typedef __attribute__((ext_vector_type(2))) float v2f;
typedef __attribute__((ext_vector_type(8))) float v8f;

// Order-preserving fp32 -> u32 key (ascending)
__device__ __forceinline__ unsigned key_of(float f) {
  unsigned u = __float_as_uint(f);
  return (u & 0x80000000u) ? ~u : (u | 0x80000000u);
}

// Wave(32)-wide sum using V_WMMA_F32_16X16X4_F32 with an all-ones A matrix.
// D[m][n] = sum_k B[k][n]  (rows identical since A is all ones), so c[0] holds
// the 16 column sums (lanes 0-15 and 16-31 mirrored); 4 shfl_xor finish it.
__device__ __forceinline__ float wave_sum_wmma(float x) {
  v2f a; a[0] = 1.0f; a[1] = 1.0f;     // A = ones(16x4)
  v2f b; b[0] = x;    b[1] = 0.0f;     // one partial per lane somewhere in B(4x16)
  v8f c = {};
  c = __builtin_amdgcn_wmma_f32_16x16x4_f32(false, a, false, b, (short)0, c, false, false);
  float r = c[0];
  r += __shfl_xor(r, 1, 32);
  r += __shfl_xor(r, 2, 32);
  r += __shfl_xor(r, 4, 32);
  r += __shfl_xor(r, 8, 32);
  return r;  // every lane holds the wave total
}

__global__ void k_zero(unsigned* __restrict__ ws, int nwords) {
  int i = blockIdx.x * blockDim.x + threadIdx.x;
  if (i < nwords) ws[i] = 0u;
}

// Pass 1: top-11-bit histogram of y_true keys + sum / sumsq partials of y_pred.
__global__ void k_pass1(const float* __restrict__ yt, const float* __restrict__ yp,
                        unsigned* __restrict__ ws, int n4, int n) {
  __shared__ unsigned hist[2048];
  __shared__ float lsA[8], lsB[8];
  for (int i = threadIdx.x; i < 2048; i += TPB) hist[i] = 0u;
  __syncthreads();

  float s = 0.f, s2 = 0.f;
  const float4* yt4 = (const float4*)yt;
  const float4* yp4 = (const float4*)yp;
  int stride = gridDim.x * TPB;
  for (int i = blockIdx.x * TPB + threadIdx.x; i < n4; i += stride) {
    float4 t = yt4[i];
    atomicAdd(&hist[key_of(t.x) >> 21], 1u);
    atomicAdd(&hist[key_of(t.y) >> 21], 1u);
    atomicAdd(&hist[key_of(t.z) >> 21], 1u);
    atomicAdd(&hist[key_of(t.w) >> 21], 1u);
    float4 p = yp4[i];
    s  += p.x + p.y + p.z + p.w;
    s2 += p.x * p.x + p.y * p.y + p.z * p.z + p.w * p.w;
  }
  if (blockIdx.x == 0 && threadIdx.x == 0) {        // scalar tail (n % 4)
    for (int i = n4 * 4; i < n; ++i) {
      atomicAdd(&hist[key_of(yt[i]) >> 21], 1u);
      float p = yp[i]; s += p; s2 += p * p;
    }
  }
  __syncthreads();
  unsigned* gh = ws + OFF_H1;
  for (int i = threadIdx.x; i < 2048; i += TPB) {
    unsigned v = hist[i];
    if (v) atomicAdd(&gh[i], v);
  }

  float wa = wave_sum_wmma(s);
  float wb = wave_sum_wmma(s2);
  int wave = threadIdx.x >> 5;
  if ((threadIdx.x & 31) == 0) { lsA[wave] = wa; lsB[wave] = wb; }
  __syncthreads();
  if (threadIdx.x == 0) {
    float a = 0.f, b = 0.f;
    for (int i = 0; i < TPB / 32; ++i) { a += lsA[i]; b += lsB[i]; }
    ((float*)ws)[OFF_SUMP + blockIdx.x]  = a;
    ((float*)ws)[OFF_SUMP2 + blockIdx.x] = b;
  }
}

__global__ void k_scan1(unsigned* __restrict__ ws,
                        unsigned r0, unsigned r1, unsigned r2, unsigned r3) {
  if (threadIdx.x == 0) {
    unsigned ranks[4] = {r0, r1, r2, r3};
    const unsigned* h = ws + OFF_H1;
    for (int q = 0; q < 4; ++q) {
      unsigned cum = 0;
      for (int b = 0; b < 2048; ++b) {
        unsigned c = h[b];
        if (ranks[q] < cum + c) { ws[OFF_SEL1 + q] = (unsigned)b; ws[OFF_REM1 + q] = ranks[q] - cum; break; }
        cum += c;
      }
    }
  }
}

// Pass 2: middle-11-bit histograms restricted to each rank's selected top bin.
__global__ void k_hist2(const float* __restrict__ yt, unsigned* __restrict__ ws, int n4, int n) {
  __shared__ unsigned h[4 * 2048];
  for (int i = threadIdx.x; i < 4 * 2048; i += TPB) h[i] = 0u;
  __syncthreads();
  unsigned s0 = ws[OFF_SEL1 + 0], s1 = ws[OFF_SEL1 + 1];
  unsigned s2 = ws[OFF_SEL1 + 2], s3 = ws[OFF_SEL1 + 3];

  const float4* yt4 = (const float4*)yt;
  int stride = gridDim.x * TPB;
  for (int i = blockIdx.x * TPB + threadIdx.x; i < n4; i += stride) {
    float4 t = yt4[i];
    float v[4] = {t.x, t.y, t.z, t.w};
    #pragma unroll
    for (int j = 0; j < 4; ++j) {
      unsigned key = key_of(v[j]);
      unsigned top = key >> 21, mid = (key >> 10) & 2047u;
      if (top == s0) atomicAdd(&h[mid], 1u);
      if (top == s1) atomicAdd(&h[2048 + mid], 1u);
      if (top == s2) atomicAdd(&h[4096 + mid], 1u);
      if (top == s3) atomicAdd(&h[6144 + mid], 1u);
    }
  }
  if (blockIdx.x == 0 && threadIdx.x == 0) {
    for (int i = n4 * 4; i < n; ++i) {
      unsigned key = key_of(yt[i]);
      unsigned top = key >> 21, mid = (key >> 10) & 2047u;
      if (top == s0) atomicAdd(&h[mid], 1u);
      if (top == s1) atomicAdd(&h[2048 + mid], 1u);
      if (top == s2) atomicAdd(&h[4096 + mid], 1u);
      if (top == s3) atomicAdd(&h[6144 + mid], 1u);
    }
  }
  __syncthreads();
  unsigned* gh = ws + OFF_H2;
  for (int i = threadIdx.x; i < 4 * 2048; i += TPB) {
    unsigned v = h[i];
    if (v) atomicAdd(&gh[i], v);
  }
}

__global__ void k_scan2(unsigned* __restrict__ ws) {
  if (threadIdx.x == 0) {
    for (int q = 0; q < 4; ++q) {
      unsigned target = ws[OFF_REM1 + q];
      const unsigned* h = ws + OFF_H2 + q * 2048;
      unsigned cum = 0;
      for (int b = 0; b < 2048; ++b) {
        unsigned c = h[b];
        if (target < cum + c) { ws[OFF_SEL2 + q] = (unsigned)b; ws[OFF_REM2 + q] = target - cum; break; }
        cum += c;
      }
    }
  }
}

// Pass 3: low-10-bit histograms restricted to each rank's 22-bit prefix.
__global__ void k_hist3(const float* __restrict__ yt, unsigned* __restrict__ ws, int n4, int n) {
  __shared__ unsigned h[4 * 1024];
  for (int i = threadIdx.x; i < 4 * 1024; i += TPB) h[i] = 0u;
  __syncthreads();
  unsigned p0 = (ws[OFF_SEL1 + 0] << 11) | ws[OFF_SEL2 + 0];
  unsigned p1 = (ws[OFF_SEL1 + 1] << 11) | ws[OFF_SEL2 + 1];
  unsigned p2 = (ws[OFF_SEL1 + 2] << 11) | ws[OFF_SEL2 + 2];
  unsigned p3 = (ws[OFF_SEL1 + 3] << 11) | ws[OFF_SEL2 + 3];

  const float4* yt4 = (const float4*)yt;
  int stride = gridDim.x * TPB;
  for (int i = blockIdx.x * TPB + threadIdx.x; i < n4; i += stride) {
    float4 t = yt4[i];
    float v[4] = {t.x, t.y, t.z, t.w};
    #pragma unroll
    for (int j = 0; j < 4; ++j) {
      unsigned key = key_of(v[j]);
      unsigned hi = key >> 10, lo = key & 1023u;
      if (hi == p0) atomicAdd(&h[lo], 1u);
      if (hi == p1) atomicAdd(&h[1024 + lo], 1u);
      if (hi == p2) atomicAdd(&h[2048 + lo], 1u);
      if (hi == p3) atomicAdd(&h[3072 + lo], 1u);
    }
  }
  if (blockIdx.x == 0 && threadIdx.x == 0) {
    for (int i = n4 * 4; i < n; ++i) {
      unsigned key = key_of(yt[i]);
      unsigned hi = key >> 10, lo = key & 1023u;
      if (hi == p0) atomicAdd(&h[lo], 1u);
      if (hi == p1) atomicAdd(&h[1024 + lo], 1u);
      if (hi == p2) atomicAdd(&h[2048 + lo], 1u);
      if (hi == p3) atomicAdd(&h[3072 + lo], 1u);
    }
  }
  __syncthreads();
  unsigned* gh = ws + OFF_H3;
  for (int i = threadIdx.x; i < 4 * 1024; i += TPB) {
    unsigned v = h[i];
    if (v) atomicAdd(&gh[i], v);
  }
}

__global__ void k_scan3(unsigned* __restrict__ ws, float fL, float fR) {
  if (threadIdx.x == 0) {
    float vals[4];
    for (int q = 0; q < 4; ++q) {
      unsigned target = ws[OFF_REM2 + q];
      unsigned prefix = (ws[OFF_SEL1 + q] << 11) | ws[OFF_SEL2 + q];
      const unsigned* h = ws + OFF_H3 + q * 1024;
      unsigned cum = 0, b3 = 0;
      for (int b = 0; b < 1024; ++b) {
        unsigned c = h[b];
        if (target < cum + c) { b3 = (unsigned)b; break; }
        cum += c;
      }
      unsigned key = (prefix << 10) | b3;
      unsigned u = (key & 0x80000000u) ? (key ^ 0x80000000u) : ~key;  // inverse transform
      vals[q] = __uint_as_float(u);
    }
    float* f = (float*)ws;
    f[OFF_VALS + 0] = vals[0]; f[OFF_VALS + 1] = vals[1];
    f[OFF_VALS + 2] = vals[2]; f[OFF_VALS + 3] = vals[3];
    f[OFF_THR + 0] = vals[0] + fL * (vals[1] - vals[0]);   // left threshold
    f[OFF_THR + 1] = vals[2] + fR * (vals[3] - vals[2]);   // right threshold
  }
}

// Final streaming pass: penalty-weighted squared residual partial sums.
__global__ void k_mse(const float* __restrict__ yt, const float* __restrict__ yp,
                      unsigned* __restrict__ ws, int n4, int n) {
  __shared__ float lsA[8];
  const float* thr = (const float*)(ws + OFF_THR);
  float tL = thr[0], tR = thr[1];

  float acc = 0.f;
  const float4* yt4 = (const float4*)yt;
  const float4* yp4 = (const float4*)yp;
  int stride = gridDim.x * TPB;
  for (int i = blockIdx.x * TPB + threadIdx.x; i < n4; i += stride) {
    float4 t = yt4[i];
    float4 p = yp4[i];
    float tv[4] = {t.x, t.y, t.z, t.w};
    float pv[4] = {p.x, p.y, p.z, p.w};
    #pragma unroll
    for (int j = 0; j < 4; ++j) {
      float w = (tv[j] < tL) ? 6.0f : ((tv[j] > tR) ? 6.0f : 1.0f);
      float r = tv[j] - pv[j];
      acc += w * r * r;
    }
  }
  if (blockIdx.x == 0 && threadIdx.x == 0) {
    for (int i = n4 * 4; i < n; ++i) {
      float tv = yt[i];
      float w = (tv < tL) ? 6.0f : ((tv > tR) ? 6.0f : 1.0f);
      float r = tv - yp[i];
      acc += w * r * r;
    }
  }

  float wsum = wave_sum_wmma(acc);
  int wave = threadIdx.x >> 5;
  if ((threadIdx.x & 31) == 0) lsA[wave] = wsum;
  __syncthreads();
  if (threadIdx.x == 0) {
    float a = 0.f;
    for (int i = 0; i < TPB / 32; ++i) a += lsA[i];
    ((float*)ws)[OFF_MSE + blockIdx.x] = a;
  }
}

// Deterministic fixed-order final combine in fp64.
__global__ void k_combine(const unsigned* __restrict__ ws, float* __restrict__ out,
                          int nblk, double n) {
  __shared__ double A[TPB], B[TPB], C[TPB];
  const float* sp  = (const float*)(ws + OFF_SUMP);
  const float* sp2 = (const float*)(ws + OFF_SUMP2);
  const float* ms  = (const float*)(ws + OFF_MSE);
  double a = 0.0, b = 0.0, c = 0.0;
  for (int i = threadIdx.x; i < nblk; i += TPB) { a += sp[i]; b += sp2[i]; c += ms[i]; }
  A[threadIdx.x] = a; B[threadIdx.x] = b; C[threadIdx.x] = c;
  __syncthreads();
  if (threadIdx.x == 0) {
    double sa = 0.0, sb = 0.0, sc = 0.0;
    for (int i = 0; i < TPB; ++i) { sa += A[i]; sb += B[i]; sc += C[i]; }
    double mse = sc / n;
    double var = (sb - sa * sa / n) / (n - 1.0);   // unbiased (ddof=1)
    out[0] = (float)(mse - var);
  }
}

extern "C" void kernel_launch(void* const* d_in, const int* in_sizes, int n_in,
                              void* d_out, int out_size, void* d_ws, size_t ws_size,
                              hipStream_t stream) {
  const float* y_pred = (const float*)d_in[0];
  const float* y_true = (const float*)d_in[1];
  float* out = (float*)d_out;
  unsigned* ws = (unsigned*)d_ws;

  int n = in_sizes[0];
  int n4 = n >> 2;

  // percentile ranks (0-based) and interpolation fractions, matching jnp.percentile
  double qL = 0.15 * (double)(n - 1);
  double qR = 0.85 * (double)(n - 1);
  unsigned kL = (unsigned)qL; float fL = (float)(qL - (double)kL);
  unsigned kR = (unsigned)qR; float fR = (float)(qR - (double)kR);

  k_zero<<<(ZERO_WORDS + TPB - 1) / TPB, TPB, 0, stream>>>(ws, ZERO_WORDS);
  k_pass1<<<NBLK, TPB, 0, stream>>>(y_true, y_pred, ws, n4, n);
  k_scan1<<<1, 32, 0, stream>>>(ws, kL, kL + 1, kR, kR + 1);
  k_hist2<<<NBLK, TPB, 0, stream>>>(y_true, ws, n4, n);
  k_scan2<<<1, 32, 0, stream>>>(ws);
  k_hist3<<<NBLK, TPB, 0, stream>>>(y_true, ws, n4, n);
  k_scan3<<<1, 32, 0, stream>>>(ws, fL, fR);
  k_mse<<<NBLK, TPB, 0, stream>>>(y_true, y_pred, ws, n4, n);
  k_combine<<<1, TPB, 0, stream>>>(ws, out, NBLK, (double)n);
}